// ConstrainedAttentionModel_20864951124645
// MI455X (gfx1250) — compile-verified
//
#include <hip/hip_runtime.h>
#include <cstdint>

typedef __attribute__((ext_vector_type(2))) float v2f;
typedef __attribute__((ext_vector_type(8))) float v8f;

#define SEQ_T 2048
#define VOCAB 8192
#define NTHR  256

__global__ __launch_bounds__(NTHR)
void ConstrainedAttentionModel_kernel(const int* __restrict__ x,
                                      const float* __restrict__ params,
                                      float* __restrict__ out)
{
    __shared__ __align__(16) int   sh_tok[SEQ_T];   // 8 KB token row
    __shared__ __align__(16) float sh_out[VOCAB];   // 32 KB scatter accumulator
    __shared__ float sh_red[16];                    // block reduction scratch

    const int b    = blockIdx.x;
    const int tid  = threadIdx.x;
    const int lane = tid & 31;
    const int wid  = tid >> 5;

    const int* xb = x + b * SEQ_T;

    // ---- gfx1250 async global->LDS DMA of the token row (ASYNCcnt path) ----
    // Each thread copies 32 bytes (two b128 transfers).
    unsigned lds_base = (unsigned)(uintptr_t)(&sh_tok[0]);
    unsigned boff     = (unsigned)tid * 32u;
    asm volatile("global_load_async_to_lds_b128 %0, %1, %2"
                 :: "v"(lds_base + boff), "v"(boff), "s"(xb) : "memory");
    asm volatile("global_load_async_to_lds_b128 %0, %1, %2"
                 :: "v"(lds_base + boff + 16u), "v"(boff + 16u), "s"(xb) : "memory");

    // Overlap: zero the LDS output accumulator while the DMA is in flight.
    #pragma unroll
    for (int i = 0; i < VOCAB / NTHR; ++i)
        sh_out[tid + i * NTHR] = 0.0f;

    asm volatile("s_wait_asynccnt 0" ::: "memory");
    __syncthreads();

    const float p0 = params[0], p1 = params[1], p2 = params[2], p3 = params[3];
    const int last  = sh_tok[SEQ_T - 1];
    const int last2 = sh_tok[SEQ_T - 2];

    // ---- scores (the one-hot einsums degenerate to integer equality) ----
    float sc[8];
    int   tok[8];
    const int t0 = tid * 8;
    #pragma unroll
    for (int i = 0; i < 8; ++i) {
        const int t    = t0 + i;
        const int cur  = sh_tok[t];
        const int prev = (t > 0) ? sh_tok[t - 1] : -1;  // t=0: zero vector, never matches
        tok[i] = cur;
        float s = 0.0f;
        s += (cur  == last)  ? p0 : 0.0f;
        s += (prev == last)  ? p1 : 0.0f;
        s += (cur  == last2) ? p2 : 0.0f;
        s += (prev == last2) ? p3 : 0.0f;
        if (t == SEQ_T - 1) s = -1000000000.0f;         // exact reference mask
        sc[i] = s;
    }

    // ---- softmax max: wave32 shfl tree + cross-wave via LDS ----
    float lmax = sc[0];
    #pragma unroll
    for (int i = 1; i < 8; ++i) lmax = fmaxf(lmax, sc[i]);
    #pragma unroll
    for (int o = 16; o > 0; o >>= 1) lmax = fmaxf(lmax, __shfl_xor(lmax, o, 32));
    if (lane == 0) sh_red[wid] = lmax;
    __syncthreads();
    float m = sh_red[0];
    #pragma unroll
    for (int i = 1; i < 8; ++i) m = fmaxf(m, sh_red[i]);

    // ---- exp + denominator ----
    float e[8];
    float lsum = 0.0f;
    #pragma unroll
    for (int i = 0; i < 8; ++i) { e[i] = expf(sc[i] - m); lsum += e[i]; }

    // Wave-wide sum of `lsum` via v_wmma_f32_16x16x4_f32:
    //   A (16x4, documented layout): VGPR0 holds K=0 (lanes 0-15) / K=2 (lanes 16-31)
    //   = per-lane partial; VGPR1 (K=1/K=3) = 0. B = all-ones (layout-invariant).
    //   D[r] per lane = rowsum(M) => sum of 8 D regs + shfl_xor(16) = full 32-lane sum.
    v2f a;  a[0] = lsum;  a[1] = 0.0f;
    v2f bo; bo[0] = 1.0f; bo[1] = 1.0f;
    v8f c = {};
    v8f d = __builtin_amdgcn_wmma_f32_16x16x4_f32(false, a, false, bo,
                                                  (short)0, c, false, false);
    float wsum = d[0] + d[1] + d[2] + d[3] + d[4] + d[5] + d[6] + d[7];
    wsum += __shfl_xor(wsum, 16, 32);
    if (lane == 0) sh_red[8 + wid] = wsum;
    __syncthreads();
    float total = sh_red[8];
    #pragma unroll
    for (int i = 1; i < 8; ++i) total += sh_red[8 + i];

    // ---- scatter attn mass onto token ids (ds_add_f32 atomics in LDS) ----
    #pragma unroll
    for (int i = 0; i < 8; ++i)
        atomicAdd(&sh_out[tok[i]], e[i] / total);
    __syncthreads();

    // ---- stream the 32 KB row out, 128b stores ----
    const float4* so4 = (const float4*)sh_out;
    float4*       o4  = (float4*)(out + b * VOCAB);
    #pragma unroll
    for (int i = 0; i < (VOCAB / 4) / NTHR; ++i)
        o4[tid + i * NTHR] = so4[tid + i * NTHR];
}

extern "C" void kernel_launch(void* const* d_in, const int* in_sizes, int n_in,
                              void* d_out, int out_size, void* d_ws, size_t ws_size,
                              hipStream_t stream) {
    (void)n_in; (void)out_size; (void)d_ws; (void)ws_size;
    const int*   x      = (const int*)d_in[0];
    const float* params = (const float*)d_in[1];
    float*       out    = (float*)d_out;
    const int B = in_sizes[0] / SEQ_T;   // 8
    ConstrainedAttentionModel_kernel<<<B, NTHR, 0, stream>>>(x, params, out);
}